// SOTALocationPredictor_53558242181867
// MI455X (gfx1250) — compile-verified
//
// SOTALocationPredictor forward for MI455X (gfx1250), fp32 WMMA 16x16x4 path.
//
// Roofline: ~50 GFLOP total, ~200MB HBM traffic @ 23.3 TB/s -> memory/launch
// bound. Keep fp32 (reference numerics: -1e30 masks, LN, erf-GELU) and fuse:
//   k_embed : gather + concat(114) + proj + LN + pos  -> x [245760,64]
//   k_attn  : per-batch fused MHA (QKV,S,softmax,PV,out-proj,res,LN) in LDS
//   k_moe   : per-16-token fused MoE (router top2, 3 experts x 2 GEMM + GELU,
//             gated combine, res, LN) in LDS
//   k_final/k_pnorm/k_head : final LN+norm, proto norm, fused 3-GEMM logits
// Input order assumes jax pytree (sorted dict key) flattening of setup_inputs().

#include <hip/hip_runtime.h>

typedef float v2f __attribute__((ext_vector_type(2)));
typedef float v8f __attribute__((ext_vector_type(8)));

#define kB   4096
#define kL   60
#define kD   64
#define kE   3
#define kV   1187
#define kTok (kB * kL)

static __device__ __forceinline__ v8f wmma_f32(v2f a, v2f b, v8f c) {
  // V_WMMA_F32_16X16X4_F32: D = A(16x4) x B(4x16) + C(16x16), wave32
  return __builtin_amdgcn_wmma_f32_16x16x4_f32(false, a, false, b, (short)0, c,
                                               false, false);
}
static __device__ __forceinline__ v8f z8() {
  v8f z = {0.f, 0.f, 0.f, 0.f, 0.f, 0.f, 0.f, 0.f};
  return z;
}
static __device__ __forceinline__ float gelu_exact(float v) {
  return 0.5f * v * (1.0f + erff(v * 0.7071067811865476f));
}

// ---------------------------------------------------------------- embed+proj
__global__ __launch_bounds__(64) void k_embed(
    const int* __restrict__ loc_seq, const int* __restrict__ user_seq,
    const int* __restrict__ weekday_seq, const int* __restrict__ start_min_seq,
    const int* __restrict__ dur_seq, const int* __restrict__ diff_seq,
    const float* __restrict__ loc_emb, const float* __restrict__ user_emb,
    const float* __restrict__ weekday_emb, const float* __restrict__ hour_emb,
    const float* __restrict__ proj_w, const float* __restrict__ proj_b,
    const float* __restrict__ ln_g, const float* __restrict__ ln_b,
    const float* __restrict__ pos_emb, float* __restrict__ x) {
  const int t = blockIdx.x;
  const int b = t / kL, l = t - b * kL;
  const int d = threadIdx.x;
  __shared__ float c[114];
  __shared__ float red[64];
  __shared__ float st[2];
  const int loc = loc_seq[t];
  c[d] = loc_emb[loc * 64 + d];
  if (d < 24) {
    const int u = user_seq[b * kL];  // user_seq[:, 0]
    c[64 + d] = user_emb[u * 24 + d];
  }
  if (d < 12) {
    const int wd = weekday_seq[t];
    c[88 + d] = weekday_emb[wd * 12 + d];
    int hr = start_min_seq[t] / 60;
    hr = hr < 0 ? 0 : (hr > 23 ? 23 : hr);
    c[100 + d] = hour_emb[hr * 12 + d];
  }
  if (d == 0) {
    c[112] = (float)diff_seq[t] * 0.1f;
    c[113] = (float)dur_seq[t] * (1.0f / 300.0f);
  }
  __syncthreads();
  float acc = proj_b[d];
  const float* wr = proj_w + d * 114;
  for (int k = 0; k < 114; ++k) acc = fmaf(wr[k], c[k], acc);
  red[d] = acc;
  __syncthreads();
  if (d == 0) {
    float m = 0.f;
    for (int i = 0; i < 64; ++i) m += red[i];
    m *= (1.0f / 64.0f);
    float v = 0.f;
    for (int i = 0; i < 64; ++i) {
      const float df = red[i] - m;
      v = fmaf(df, df, v);
    }
    st[0] = m;
    st[1] = rsqrtf(v * (1.0f / 64.0f) + 1e-5f);
  }
  __syncthreads();
  x[(size_t)t * kD + d] =
      (acc - st[0]) * st[1] * ln_g[d] + ln_b[d] + pos_emb[l * kD + d];
}

// --------------------------------------------------- fused attention block
// One block per batch row b (4 waves, 128 thr). Wave w owns token rows
// 16w..16w+15 (L padded to 64 with zero rows). Everything staged in LDS;
// x updated in place with LN(x + attn_out).
__global__ __launch_bounds__(128) void k_attn(
    float* __restrict__ x, const int* __restrict__ loc_seq,
    const float* __restrict__ in_w, const float* __restrict__ in_b,
    const float* __restrict__ out_w, const float* __restrict__ out_b,
    const float* __restrict__ n1_g, const float* __restrict__ n1_b) {
  __shared__ float Xs[64 * 65];
  __shared__ float Qs[64 * 17];
  __shared__ float Ks[64 * 17];
  __shared__ float Vs[64 * 17];
  __shared__ float Ss[64 * 65];
  __shared__ float Os[64 * 65];
  __shared__ float rmean[64], rstd[64];

  const int b = blockIdx.x;
  const int tid = threadIdx.x;
  const int wv = tid >> 5;
  const int lane = tid & 31;
  const int l16 = lane & 15;
  const int hi = lane >> 4;
  const int ko = hi * 2;  // K offset within A/B frag
  const int ro = hi * 8;  // row offset within C/D frag
  const int m0 = wv * 16;
  float* xb = x + (size_t)b * kL * kD;

  for (int i = tid; i < 64 * 64; i += 128) {
    const int r = i >> 6, cc = i & 63;
    Xs[r * 65 + cc] = (r < kL) ? xb[r * kD + cc] : 0.f;
  }
  __syncthreads();

  for (int h = 0; h < 4; ++h) {
    // Q/K/V strips for this wave's 16 rows: [16x16] = Xs[16x64] @ W^T
    for (int sel = 0; sel < 3; ++sel) {
      const float* W = in_w + (sel * 64 + h * 16) * 64;
      v8f acc = z8();
#pragma unroll
      for (int k0 = 0; k0 < 64; k0 += 4) {
        v2f a, bf;
        const float* ar = &Xs[(m0 + l16) * 65 + k0 + ko];
        a.x = ar[0];
        a.y = ar[1];
        const float* br = W + l16 * 64 + k0 + ko;
        bf.x = br[0];
        bf.y = br[1];
        acc = wmma_f32(a, bf, acc);
      }
      const float bias = in_b[sel * 64 + h * 16 + l16];
      float* dst = (sel == 0) ? Qs : (sel == 1) ? Ks : Vs;
#pragma unroll
      for (int i = 0; i < 8; ++i) dst[(m0 + ro + i) * 17 + l16] = acc[i] + bias;
    }
    __syncthreads();

    // S strip [16x64] = Q strip @ K^T, scale, key-pad mask
#pragma unroll
    for (int nt = 0; nt < 4; ++nt) {
      v8f acc = z8();
#pragma unroll
      for (int k0 = 0; k0 < 16; k0 += 4) {
        v2f a, bf;
        const float* ar = &Qs[(m0 + l16) * 17 + k0 + ko];
        a.x = ar[0];
        a.y = ar[1];
        const float* br = &Ks[(nt * 16 + l16) * 17 + k0 + ko];
        bf.x = br[0];
        bf.y = br[1];
        acc = wmma_f32(a, bf, acc);
      }
      const int col = nt * 16 + l16;
      const bool pad = col >= kL;
      const bool msk = (!pad) && (loc_seq[b * kL + col] == 0);
#pragma unroll
      for (int i = 0; i < 8; ++i) {
        float v = acc[i] * 0.25f;  // 1/sqrt(HD=16)
        v = pad ? -1e38f : (msk ? -1e30f : v);
        Ss[(m0 + ro + i) * 65 + col] = v;
      }
    }
    // softmax over 64 cols (padded cols -> exp==0); wave-private strip
    if (lane < 16) {
      float* row = &Ss[(m0 + lane) * 65];
      float mx = -3.4e38f;
      for (int cc = 0; cc < 64; ++cc) mx = fmaxf(mx, row[cc]);
      float sum = 0.f;
      for (int cc = 0; cc < 64; ++cc) {
        const float e = expf(row[cc] - mx);
        sum += e;
        row[cc] = e;
      }
      const float inv = 1.0f / sum;
      for (int cc = 0; cc < 64; ++cc) row[cc] *= inv;
    }
    // O strip [16x16] = P[16x64] @ V[64x16]
    {
      v8f acc = z8();
#pragma unroll
      for (int k0 = 0; k0 < 64; k0 += 4) {
        v2f a, bf;
        const float* ar = &Ss[(m0 + l16) * 65 + k0 + ko];
        a.x = ar[0];
        a.y = ar[1];
        bf.x = Vs[(k0 + ko) * 17 + l16];
        bf.y = Vs[(k0 + ko + 1) * 17 + l16];
        acc = wmma_f32(a, bf, acc);
      }
#pragma unroll
      for (int i = 0; i < 8; ++i)
        Os[(m0 + ro + i) * 65 + h * 16 + l16] = acc[i];
    }
    __syncthreads();  // protect Q/K/V tiles for next head
  }

  // out-proj [16x64] + bias + residual -> stage in Ss
#pragma unroll
  for (int nt = 0; nt < 4; ++nt) {
    v8f acc = z8();
#pragma unroll
    for (int k0 = 0; k0 < 64; k0 += 4) {
      v2f a, bf;
      const float* ar = &Os[(m0 + l16) * 65 + k0 + ko];
      a.x = ar[0];
      a.y = ar[1];
      const float* br = out_w + (nt * 16 + l16) * 64 + k0 + ko;
      bf.x = br[0];
      bf.y = br[1];
      acc = wmma_f32(a, bf, acc);
    }
    const float bias = out_b[nt * 16 + l16];
#pragma unroll
    for (int i = 0; i < 8; ++i) {
      const int r = m0 + ro + i, cc = nt * 16 + l16;
      Ss[r * 65 + cc] = acc[i] + bias + Xs[r * 65 + cc];
    }
  }
  if (lane < 16) {
    const int r = m0 + lane;
    const float* row = &Ss[r * 65];
    float m = 0.f;
    for (int cc = 0; cc < 64; ++cc) m += row[cc];
    m *= (1.0f / 64.0f);
    float v = 0.f;
    for (int cc = 0; cc < 64; ++cc) {
      const float df = row[cc] - m;
      v = fmaf(df, df, v);
    }
    rmean[r] = m;
    rstd[r] = rsqrtf(v * (1.0f / 64.0f) + 1e-5f);
  }
  __syncthreads();
  for (int i = tid; i < kL * 64; i += 128) {
    const int r = i >> 6, cc = i & 63;
    xb[r * kD + cc] = (Ss[r * 65 + cc] - rmean[r]) * rstd[r] * n1_g[cc] + n1_b[cc];
  }
}

// -------------------------------------------------------------- fused MoE
// One wave per 16-token strip; x updated in place with LN(x + moe).
__global__ __launch_bounds__(32) void k_moe(
    float* __restrict__ x, const float* __restrict__ router_w,
    const float* __restrict__ w1, const float* __restrict__ b1,
    const float* __restrict__ w2, const float* __restrict__ b2,
    const float* __restrict__ n2_g, const float* __restrict__ n2_b) {
  __shared__ float Xs[16 * 65];
  __shared__ float Hs[16 * 65];
  __shared__ float Ys[16 * 65];
  __shared__ float gl[16 * kE];
  __shared__ float rm[16], rs_[16];
  const int lane = threadIdx.x;
  const int l16 = lane & 15;
  const int hi = lane >> 4;
  const int ko = hi * 2, ro = hi * 8;
  float* xb = x + (size_t)blockIdx.x * 16 * kD;

  for (int i = lane; i < 16 * 64; i += 32) {
    const int r = i >> 6, cc = i & 63;
    Xs[r * 65 + cc] = xb[i];
  }
  __syncthreads();

  // router softmax(3) + top-2 renormalized gates (tie rule matches lax.top_k)
  if (lane < 16) {
    const float* xr = &Xs[lane * 65];
    float lg[3];
#pragma unroll
    for (int e = 0; e < kE; ++e) {
      float s = 0.f;
      for (int k = 0; k < kD; ++k) s = fmaf(router_w[e * kD + k], xr[k], s);
      lg[e] = s;
    }
    const float m = fmaxf(lg[0], fmaxf(lg[1], lg[2]));
    float rw[3] = {expf(lg[0] - m), expf(lg[1] - m), expf(lg[2] - m)};
    const float tot = rw[0] + rw[1] + rw[2];
    rw[0] /= tot;
    rw[1] /= tot;
    rw[2] /= tot;
    int j = 0;  // dropped expert: argmin, larger index wins ties
    if (rw[1] <= rw[j]) j = 1;
    if (rw[2] <= rw[j]) j = 2;
    const float keep = rw[0] + rw[1] + rw[2] - rw[j];
#pragma unroll
    for (int e = 0; e < kE; ++e) gl[lane * kE + e] = (e == j) ? 0.f : rw[e] / keep;
  }
  __syncthreads();

  v8f yacc[4] = {z8(), z8(), z8(), z8()};
  for (int e = 0; e < kE; ++e) {
    // H = gelu(X @ w1[e]^T + b1[e])
#pragma unroll
    for (int nt = 0; nt < 4; ++nt) {
      v8f acc = z8();
#pragma unroll
      for (int k0 = 0; k0 < 64; k0 += 4) {
        v2f a, bf;
        const float* ar = &Xs[l16 * 65 + k0 + ko];
        a.x = ar[0];
        a.y = ar[1];
        const float* br = w1 + ((e * 64 + nt * 16 + l16) * 64 + k0 + ko);
        bf.x = br[0];
        bf.y = br[1];
        acc = wmma_f32(a, bf, acc);
      }
      const float bias = b1[e * 64 + nt * 16 + l16];
#pragma unroll
      for (int i = 0; i < 8; ++i)
        Hs[(ro + i) * 65 + nt * 16 + l16] = gelu_exact(acc[i] + bias);
    }
    __syncthreads();
    // y += g_e * (H @ w2[e]^T + b2[e])
#pragma unroll
    for (int nt = 0; nt < 4; ++nt) {
      v8f acc = z8();
#pragma unroll
      for (int k0 = 0; k0 < 64; k0 += 4) {
        v2f a, bf;
        const float* ar = &Hs[l16 * 65 + k0 + ko];
        a.x = ar[0];
        a.y = ar[1];
        const float* br = w2 + ((e * 64 + nt * 16 + l16) * 64 + k0 + ko);
        bf.x = br[0];
        bf.y = br[1];
        acc = wmma_f32(a, bf, acc);
      }
      const float bias = b2[e * 64 + nt * 16 + l16];
#pragma unroll
      for (int i = 0; i < 8; ++i)
        yacc[nt][i] += gl[(ro + i) * kE + e] * (acc[i] + bias);
    }
    __syncthreads();
  }

  // residual + LN in place
#pragma unroll
  for (int nt = 0; nt < 4; ++nt)
#pragma unroll
    for (int i = 0; i < 8; ++i) {
      const int r = ro + i, cc = nt * 16 + l16;
      Ys[r * 65 + cc] = yacc[nt][i] + Xs[r * 65 + cc];
    }
  if (lane < 16) {
    const float* row = &Ys[lane * 65];
    float m = 0.f;
    for (int cc = 0; cc < 64; ++cc) m += row[cc];
    m *= (1.0f / 64.0f);
    float v = 0.f;
    for (int cc = 0; cc < 64; ++cc) {
      const float df = row[cc] - m;
      v = fmaf(df, df, v);
    }
    rm[lane] = m;
    rs_[lane] = rsqrtf(v * (1.0f / 64.0f) + 1e-5f);
  }
  __syncthreads();
  for (int i = lane; i < 16 * 64; i += 32) {
    const int r = i >> 6, cc = i & 63;
    xb[i] = (Ys[r * 65 + cc] - rm[r]) * rs_[r] * n2_g[cc] + n2_b[cc];
  }
}

// ----------------------------------------------- final LN + L2 normalize
__global__ __launch_bounds__(64) void k_final(
    const float* __restrict__ x, const int* __restrict__ seq_len,
    const float* __restrict__ fn_g, const float* __restrict__ fn_b,
    float* __restrict__ finalv, float* __restrict__ fnv) {
  const int b = blockIdx.x, d = threadIdx.x;
  __shared__ float buf[64];
  __shared__ float st[2];
  int idx = seq_len[b] - 1;
  idx = idx < 0 ? 0 : (idx > kL - 1 ? kL - 1 : idx);
  const float val = x[((size_t)b * kL + idx) * kD + d];
  buf[d] = val;
  __syncthreads();
  if (d == 0) {
    float m = 0.f;
    for (int i = 0; i < 64; ++i) m += buf[i];
    m *= (1.0f / 64.0f);
    float v = 0.f;
    for (int i = 0; i < 64; ++i) {
      const float df = buf[i] - m;
      v = fmaf(df, df, v);
    }
    st[0] = m;
    st[1] = rsqrtf(v * (1.0f / 64.0f) + 1e-5f);
  }
  __syncthreads();
  const float f = (val - st[0]) * st[1] * fn_g[d] + fn_b[d];
  finalv[b * kD + d] = f;
  __syncthreads();
  buf[d] = f * f;
  __syncthreads();
  if (d == 0) {
    float s = 0.f;
    for (int i = 0; i < 64; ++i) s += buf[i];
    st[0] = fmaxf(sqrtf(s), 1e-12f);
  }
  __syncthreads();
  fnv[b * kD + d] = f / st[0];
}

__global__ __launch_bounds__(64) void k_pnorm(const float* __restrict__ protos,
                                              float* __restrict__ pn) {
  const int v = blockIdx.x, d = threadIdx.x;
  __shared__ float buf[64];
  __shared__ float nrm;
  const float val = protos[(size_t)v * kD + d];
  buf[d] = val * val;
  __syncthreads();
  if (d == 0) {
    float s = 0.f;
    for (int i = 0; i < 64; ++i) s += buf[i];
    nrm = fmaxf(sqrtf(s), 1e-12f);
  }
  __syncthreads();
  pn[(size_t)v * kD + d] = val / nrm;
}

// ------------------------------------- fused logits: cosine + 2 linear heads
__global__ __launch_bounds__(32) void k_head(
    const float* __restrict__ finalv, const float* __restrict__ fnv,
    const float* __restrict__ pn, const float* __restrict__ head_w,
    const float* __restrict__ head_b, const float* __restrict__ head_weights,
    const float* __restrict__ temp, float* __restrict__ out) {
  const int lane = threadIdx.x;
  const int l16 = lane & 15, hi = lane >> 4;
  const int ko = hi * 2, ro = hi * 8;
  const int m0 = blockIdx.x * 16;
  const int n0 = blockIdx.y * 16;
  const int n = n0 + l16;
  const bool nv = n < kV;
  v8f ac = z8(), a0 = z8(), a1 = z8();
#pragma unroll
  for (int k0 = 0; k0 < 64; k0 += 4) {
    v2f af, aF, bp, b0, b1;
    const float* fr = fnv + (m0 + l16) * kD + k0 + ko;
    af.x = fr[0];
    af.y = fr[1];
    const float* Fr = finalv + (m0 + l16) * kD + k0 + ko;
    aF.x = Fr[0];
    aF.y = Fr[1];
    if (nv) {
      const float* p = pn + (size_t)n * kD + k0 + ko;
      bp.x = p[0];
      bp.y = p[1];
      const float* h0 = head_w + (size_t)n * kD + k0 + ko;
      b0.x = h0[0];
      b0.y = h0[1];
      const float* h1 = head_w + (size_t)(kV + n) * kD + k0 + ko;
      b1.x = h1[0];
      b1.y = h1[1];
    } else {
      bp.x = bp.y = b0.x = b0.y = b1.x = b1.y = 0.f;
    }
    ac = wmma_f32(af, bp, ac);
    a0 = wmma_f32(aF, b0, a0);
    a1 = wmma_f32(aF, b1, a1);
  }
  const float w0r = head_weights[0], w1r = head_weights[1];
  const float mw = fmaxf(w0r, w1r);
  const float e0 = expf(w0r - mw), e1 = expf(w1r - mw);
  const float hw0 = e0 / (e0 + e1), hw1 = e1 / (e0 + e1);
  float tc = temp[0];
  tc = fminf(fmaxf(tc, 0.1f), 5.0f);
  if (nv) {
    const float hb0 = head_b[n], hb1 = head_b[kV + n];
#pragma unroll
    for (int i = 0; i < 8; ++i) {
      const int m = m0 + ro + i;
      const float mh = hw0 * (a0[i] + hb0) + hw1 * (a1[i] + hb1);
      out[(size_t)m * kV + n] = (6.0f * ac[i] + 0.4f * mh) / tc;
    }
  }
}

// ---------------------------------------------------------------- launcher
extern "C" void kernel_launch(void* const* d_in, const int* in_sizes, int n_in,
                              void* d_out, int out_size, void* d_ws,
                              size_t ws_size, hipStream_t stream) {
  (void)in_sizes;
  (void)out_size;
  (void)ws_size;
  if (n_in < 49) return;  // expects jax pytree (sorted-key) flatten: 49 leaves

  const int* diff_seq = (const int*)d_in[0];
  const int* dur_seq = (const int*)d_in[1];
  const int* loc_seq = (const int*)d_in[2];
  const float* fn_b = (const float*)d_in[3];
  const float* fn_g = (const float*)d_in[4];
  const float* head_b = (const float*)d_in[5];
  const float* head_w = (const float*)d_in[6];
  const float* head_weights = (const float*)d_in[7];
  const float* hour_emb = (const float*)d_in[8];
  // layers: base 9 + 13*ly:
  //  +0 attn_in_b +1 attn_in_w +2 attn_out_b +3 attn_out_w +4 b1 +5 b2
  //  +6 n1_b +7 n1_g +8 n2_b +9 n2_g +10 router_w +11 w1 +12 w2
  const float* loc_emb = (const float*)d_in[35];
  const float* pos_emb = (const float*)d_in[36];
  const float* proj_b = (const float*)d_in[37];
  const float* proj_ln_b = (const float*)d_in[38];
  const float* proj_ln_g = (const float*)d_in[39];
  const float* proj_w = (const float*)d_in[40];
  const float* protos = (const float*)d_in[41];
  const float* temp = (const float*)d_in[42];
  const float* user_emb = (const float*)d_in[43];
  const float* weekday_emb = (const float*)d_in[44];
  const int* seq_len = (const int*)d_in[45];
  const int* start_min_seq = (const int*)d_in[46];
  const int* user_seq = (const int*)d_in[47];
  const int* weekday_seq = (const int*)d_in[48];

  float* ws = (float*)d_ws;
  float* x = ws;                               // kTok*64  (62.9 MB)
  float* finalv = x + (size_t)kTok * kD;       // kB*64
  float* fnv = finalv + (size_t)kB * kD;       // kB*64
  float* pnv = fnv + (size_t)kB * kD;          // kV*64

  k_embed<<<kTok, 64, 0, stream>>>(loc_seq, user_seq, weekday_seq,
                                   start_min_seq, dur_seq, diff_seq, loc_emb,
                                   user_emb, weekday_emb, hour_emb, proj_w,
                                   proj_b, proj_ln_g, proj_ln_b, pos_emb, x);

  for (int ly = 0; ly < 2; ++ly) {
    const int base = 9 + ly * 13;
    const float* attn_in_b = (const float*)d_in[base + 0];
    const float* attn_in_w = (const float*)d_in[base + 1];
    const float* attn_out_b = (const float*)d_in[base + 2];
    const float* attn_out_w = (const float*)d_in[base + 3];
    const float* b1 = (const float*)d_in[base + 4];
    const float* b2 = (const float*)d_in[base + 5];
    const float* n1_b = (const float*)d_in[base + 6];
    const float* n1_g = (const float*)d_in[base + 7];
    const float* n2_b = (const float*)d_in[base + 8];
    const float* n2_g = (const float*)d_in[base + 9];
    const float* router_w = (const float*)d_in[base + 10];
    const float* w1 = (const float*)d_in[base + 11];
    const float* w2 = (const float*)d_in[base + 12];

    k_attn<<<kB, 128, 0, stream>>>(x, loc_seq, attn_in_w, attn_in_b,
                                   attn_out_w, attn_out_b, n1_g, n1_b);
    k_moe<<<kTok / 16, 32, 0, stream>>>(x, router_w, w1, b1, w2, b2, n2_g,
                                        n2_b);
  }

  k_final<<<kB, 64, 0, stream>>>(x, seq_len, fn_g, fn_b, finalv, fnv);
  k_pnorm<<<kV, 64, 0, stream>>>(protos, pnv);
  k_head<<<dim3(kB / 16, (kV + 15) / 16), 32, 0, stream>>>(
      finalv, fnv, pnv, head_w, head_b, head_weights, temp, (float*)d_out);
}